// EndNodeSelector_46454366273754
// MI455X (gfx1250) — compile-verified
//
#include <hip/hip_runtime.h>
#include <math.h>

typedef __attribute__((ext_vector_type(2))) float v2f;
typedef __attribute__((ext_vector_type(8))) float v8f;

#define HD 64   // hidden width (conv outputs), N dim of all GEMMs

// Flat LDS pointer -> 32-bit LDS byte offset (flat LDS addr = {aperture, offset}).
__device__ __forceinline__ unsigned lds_off(const void* p) {
    return (unsigned)(size_t)p;
}

// ---------------------------------------------------------------------------
// C[n x 64] = A[n x K] @ W[K x 64]   (row-major, K % CHUNK == 0, n % 16 == 0)
// One wave per 16-row tile (4 waves/block). Both the A tile (per-wave) and the
// W chunk (block-shared) are staged to LDS with global_load_async_to_lds_b128,
// double buffered; inner loop is pure ds_load + v_wmma_f32_16x16x4_f32.
// ldsB is declared FIRST so B-fragment ds_load offsets fit the 16-bit DS
// immediate field and fold into the instruction (no per-load v_add).
// ---------------------------------------------------------------------------
template <int CHUNK>
__global__ __launch_bounds__(128) void k_wmma_gemm(
    const float* __restrict__ A, const float* __restrict__ W,
    float* __restrict__ C, int n, int K)
{
    __shared__ __align__(16) float ldsB[2][CHUNK * HD];       // low LDS offsets
    __shared__ __align__(16) float ldsA[4][2][16 * CHUNK];
    const int wave = threadIdx.x >> 5;
    const int lane = threadIdx.x & 31;
    const int ntiles = n / 16;
    int tile = blockIdx.x * 4 + wave;           // 16-row tile (wave-uniform)
    if (tile >= ntiles) tile = ntiles - 1;      // clamp: keep all waves live for
                                                // barriers + B staging (dup stores ok)
    const int row0 = tile * 16;
    const int half = lane >> 4;                 // 0: lanes 0-15, 1: lanes 16-31
    const int lrow = lane & 15;
    const int c4   = CHUNK / 4;                 // float4 elems per row of A tile

    v8f z = {0.f,0.f,0.f,0.f,0.f,0.f,0.f,0.f};
    v8f acc[4] = {z, z, z, z};

    // Per-wave A tile: 16 rows x CHUNK floats, async global->LDS.
    auto issueA = [&](int buf, int kc) {
        for (int it = 0; it < CHUNK / 8; ++it) {
            int e = it * 32 + lane;             // flat float4 element id
            int r = e / c4, cc = e % c4;
            const float* g = A + (size_t)(row0 + r) * K + (kc + cc * 4);
            unsigned l = lds_off(&ldsA[wave][buf][e * 4]);
            unsigned long long ga = (unsigned long long)(size_t)g;
            asm volatile("global_load_async_to_lds_b128 %0, %1, off"
                         :: "v"(l), "v"(ga) : "memory");
        }
    };
    // Block-shared W chunk: CHUNK rows x 64 floats; each wave loads a quarter.
    auto issueB = [&](int buf, int kc) {
        for (int it = 0; it < CHUNK / 8; ++it) {
            int f = wave * (CHUNK * 4) + it * 32 + lane;   // float4 id in [0, CHUNK*16)
            int kr = f >> 4, cb = f & 15;
            const float* g = W + (size_t)(kc + kr) * HD + cb * 4;
            unsigned l = lds_off(&ldsB[buf][f * 4]);
            unsigned long long ga = (unsigned long long)(size_t)g;
            asm volatile("global_load_async_to_lds_b128 %0, %1, off"
                         :: "v"(l), "v"(ga) : "memory");
        }
    };

    const int nch = K / CHUNK;
    issueA(0, 0);
    issueB(0, 0);
    asm volatile("s_wait_asynccnt 0" ::: "memory");
    __syncthreads();
    for (int c = 0; c < nch; ++c) {
        const int buf = c & 1;
        if (c + 1 < nch) {                      // prefetch next chunk into other buf
            issueA(buf ^ 1, (c + 1) * CHUNK);
            issueB(buf ^ 1, (c + 1) * CHUNK);
        }
        const float* __restrict__ as = &ldsA[wave][buf][0];
        const float* __restrict__ bs = &ldsB[buf][0];
        #pragma unroll 2
        for (int kk = 0; kk < CHUNK; kk += 4) {
            // A frag 16x4 f32: VGPR0/1 = K {0,1} (lanes 0-15) / K {2,3} (lanes 16-31)
            v2f a = *(const v2f*)&as[lrow * CHUNK + kk + 2 * half];
            const int kl = kk + 2 * half;
            #pragma unroll
            for (int nt = 0; nt < 4; ++nt) {
                const int col = nt * 16 + lrow;
                v2f b;                           // B frag 4x16: rows K striped like A
                b.x = bs[kl * HD + col];
                b.y = bs[(kl + 1) * HD + col];
                acc[nt] = __builtin_amdgcn_wmma_f32_16x16x4_f32(
                    false, a, false, b, (short)0, acc[nt], false, false);
            }
        }
        if (c + 1 < nch) {
            // own prefetch done (had a whole chunk of WMMA to overlap), then
            // block barrier: c+1 visible to all waves, and all done reading buf
            asm volatile("s_wait_asynccnt 0" ::: "memory");
            __syncthreads();
        }
    }
    // D layout: VGPR r -> M = r + 8*half, N = nt*16 + lrow
    #pragma unroll
    for (int nt = 0; nt < 4; ++nt)
        #pragma unroll
        for (int r = 0; r < 8; ++r)
            C[(size_t)(row0 + r + 8 * half) * HD + nt * 16 + lrow] = acc[nt][r];
}

// ---------------------------------------------------------------------------
// Elementwise / scatter helpers
// ---------------------------------------------------------------------------
__global__ void k_fill_f32(float* p, float v, size_t cnt) {
    size_t i = (size_t)blockIdx.x * blockDim.x + threadIdx.x;
    if (i < cnt) p[i] = v;
}
__global__ void k_fill_i32(int* p, int v, int cnt) {
    int i = blockIdx.x * blockDim.x + threadIdx.x;
    if (i < cnt) p[i] = v;
}
__global__ void k_deg(const int* __restrict__ dst, int E, float* __restrict__ deg) {
    int e = blockIdx.x * blockDim.x + threadIdx.x;
    if (e < E) atomicAdd(&deg[dst[e]], 1.0f);
}
__global__ void k_dinv(const float* __restrict__ deg, float* __restrict__ dinv, int n) {
    int i = blockIdx.x * blockDim.x + threadIdx.x;
    if (i < n) dinv[i] = rsqrtf(deg[i] + 1.0f);   // +1 = self loop
}
// out[i,:] = t[i,:] * dinv[i]^2 + bias   (self-loop term + bias), thread = (i, h4)
__global__ void k_agg_init(const float* __restrict__ t, const float* __restrict__ dinv,
                           const float* __restrict__ bias, float* __restrict__ out, int n) {
    int idx = blockIdx.x * blockDim.x + threadIdx.x;
    int i = idx >> 4, h4 = idx & 15;
    if (i < n) {
        float d = dinv[i], s = d * d;
        float4 tv = ((const float4*)t)[(size_t)i * 16 + h4];
        float4 bv = ((const float4*)bias)[h4];
        float4 o  = { tv.x * s + bv.x, tv.y * s + bv.y, tv.z * s + bv.z, tv.w * s + bv.w };
        ((float4*)out)[(size_t)i * 16 + h4] = o;
    }
}
// out[dst,:] += dinv[src]*dinv[dst] * t[src,:], thread = (e, h4)
__global__ void k_agg_edges(const int* __restrict__ src, const int* __restrict__ dst,
                            const float* __restrict__ dinv, const float* __restrict__ t,
                            float* __restrict__ out, int E) {
    int idx = blockIdx.x * blockDim.x + threadIdx.x;
    int e = idx >> 4, h4 = idx & 15;
    if (e < E) {
        int s = src[e], d = dst[e];
        float nrm = dinv[s] * dinv[d];
        float4 tv = ((const float4*)t)[(size_t)s * 16 + h4];
        float* o = out + (size_t)d * HD + h4 * 4;
        atomicAdd(o + 0, nrm * tv.x);
        atomicAdd(o + 1, nrm * tv.y);
        atomicAdd(o + 2, nrm * tv.z);
        atomicAdd(o + 3, nrm * tv.w);
    }
}
__global__ void k_map_set(const int* __restrict__ smap, int M, int* __restrict__ map) {
    int j = blockIdx.x * blockDim.x + threadIdx.x;
    if (j < M) map[smap[M + j]] = smap[j];        // map[s1[j]] = s0[j]
}
// x1[ae1[e],:] = h1[map[ae0[e]],:] when mapped, thread = (e, h4)
__global__ void k_scatter_x1(const int* __restrict__ ae, const int* __restrict__ map,
                             const float* __restrict__ h1, float* __restrict__ x1, int E) {
    int idx = blockIdx.x * blockDim.x + threadIdx.x;
    int e = idx >> 4, h4 = idx & 15;
    if (e < E) {
        int g = map[ae[e]];
        if (g >= 0) {
            int tn = ae[E + e];
            ((float4*)x1)[(size_t)tn * 16 + h4] = ((const float4*)h1)[(size_t)g * 16 + h4];
        }
    }
}
__device__ __forceinline__ float eluf(float v) { return v > 0.f ? v : expf(v) - 1.f; }

__global__ void k_concat_elu(const float* __restrict__ h1, const float* __restrict__ x1,
                             float* __restrict__ hcat, int n) {
    size_t idx = (size_t)blockIdx.x * blockDim.x + threadIdx.x;
    if (idx < (size_t)n * 128) {
        size_t i = idx >> 7; int h = (int)(idx & 127);
        float v = (h < HD) ? h1[i * HD + h] : x1[i * HD + (h - HD)];
        hcat[idx] = eluf(v);
    }
}
__global__ void k_logits(const float* __restrict__ h2, const float* __restrict__ fcw,
                         const float* __restrict__ fcb, const unsigned char* __restrict__ mask,
                         float* __restrict__ logits, int n) {
    __shared__ float w[HD];
    if (threadIdx.x < HD) w[threadIdx.x] = fcw[threadIdx.x];
    __syncthreads();
    int i = blockIdx.x * blockDim.x + threadIdx.x;
    if (i < n) {
        float s = fcb[0];
        #pragma unroll 4
        for (int h = 0; h < HD; ++h) s += eluf(h2[(size_t)i * HD + h]) * w[h];
        logits[i] = mask[i] ? -1e9f : s;
    }
}
__global__ void k_max(const float* __restrict__ v, int n, float* __restrict__ red) {
    __shared__ float sm[256];
    float m = -3.4e38f;
    for (int i = threadIdx.x; i < n; i += 256) m = fmaxf(m, v[i]);
    sm[threadIdx.x] = m; __syncthreads();
    for (int s = 128; s > 0; s >>= 1) {
        if ((int)threadIdx.x < s) sm[threadIdx.x] = fmaxf(sm[threadIdx.x], sm[threadIdx.x + s]);
        __syncthreads();
    }
    if (threadIdx.x == 0) red[0] = sm[0];
}
__global__ void k_sumexp(const float* __restrict__ v, int n, float* __restrict__ red) {
    __shared__ float sm[256];
    float mx = red[0];
    float s = 0.f;
    for (int i = threadIdx.x; i < n; i += 256) s += expf(v[i] - mx);
    sm[threadIdx.x] = s; __syncthreads();
    for (int t = 128; t > 0; t >>= 1) {
        if ((int)threadIdx.x < t) sm[threadIdx.x] += sm[threadIdx.x + t];
        __syncthreads();
    }
    if (threadIdx.x == 0) red[1] = sm[0];
}
__global__ void k_final(const float* __restrict__ v, const float* __restrict__ red,
                        float* __restrict__ out, int n) {
    int i = blockIdx.x * blockDim.x + threadIdx.x;
    if (i < n) out[i] = v[i] - red[0] - logf(red[1]);
}

// ---------------------------------------------------------------------------
extern "C" void kernel_launch(void* const* d_in, const int* in_sizes, int n_in,
                              void* d_out, int out_size, void* d_ws, size_t ws_size,
                              hipStream_t stream) {
    const float* x   = (const float*)d_in[0];
    const int*  edge = (const int*)d_in[1];    // [2,E]
    const int*  aedg = (const int*)d_in[2];    // [2,E]
    const int*  smap = (const int*)d_in[3];    // [2,M]
    const unsigned char* emask = (const unsigned char*)d_in[4];
    const float* w1 = (const float*)d_in[5];
    const float* b1 = (const float*)d_in[6];
    const float* w2 = (const float*)d_in[7];
    const float* b2 = (const float*)d_in[8];
    const float* fcw = (const float*)d_in[9];
    const float* fcb = (const float*)d_in[10];
    float* out = (float*)d_out;

    const int n  = in_sizes[4];            // 50000 (e_mask)
    const int E  = in_sizes[1] / 2;        // 800000
    const int M  = in_sizes[3] / 2;        // 4096
    const int K1 = in_sizes[0] / n;        // 5000
    const int K2 = 2 * HD;                 // 128

    char* wsp = (char*)d_ws; size_t off = 0;
    auto take = [&](size_t bytes) { void* p = wsp + off; off += (bytes + 255) & ~(size_t)255; return p; };
    float* t1     = (float*)take((size_t)n * HD * 4);
    float* h1     = (float*)take((size_t)n * HD * 4);
    float* x1     = (float*)take((size_t)n * HD * 4);
    float* hcat   = (float*)take((size_t)n * K2 * 4);
    float* t2     = (float*)take((size_t)n * HD * 4);
    float* h2     = (float*)take((size_t)n * HD * 4);
    float* deg    = (float*)take((size_t)n * 4);
    float* dinv   = (float*)take((size_t)n * 4);
    int*   mapv   = (int*)  take((size_t)n * 4);
    float* logits = (float*)take((size_t)n * 4);
    float* red    = (float*)take(2 * 4);

    const int T = 256;
    auto g1 = [&](size_t cnt) { return dim3((unsigned)((cnt + T - 1) / T)); };

    // degrees / normalization (shared by both convs — same edge_index)
    k_fill_f32<<<g1(n), T, 0, stream>>>(deg, 0.f, (size_t)n);
    k_deg<<<g1(E), T, 0, stream>>>(edge + E, E, deg);
    k_dinv<<<g1(n), T, 0, stream>>>(deg, dinv, n);

    // conv1: t1 = x @ W1  (the 1 GB stream — async-staged WMMA)
    dim3 gg((unsigned)((n / 16 + 3) / 4));
    k_wmma_gemm<200><<<gg, 128, 0, stream>>>(x, w1, t1, n, K1);
    k_agg_init<<<g1((size_t)n * 16), T, 0, stream>>>(t1, dinv, b1, h1, n);
    k_agg_edges<<<g1((size_t)E * 16), T, 0, stream>>>(edge, edge + E, dinv, t1, h1, E);

    // mapping scatter: x1[ae1] = h1[map[ae0]]
    k_fill_i32<<<g1(n), T, 0, stream>>>(mapv, -1, n);
    k_map_set<<<g1(M), T, 0, stream>>>(smap, M, mapv);
    k_fill_f32<<<g1((size_t)n * HD), T, 0, stream>>>(x1, 0.f, (size_t)n * HD);
    k_scatter_x1<<<g1((size_t)E * 16), T, 0, stream>>>(aedg, mapv, h1, x1, E);

    // concat + elu -> conv2
    k_concat_elu<<<g1((size_t)n * K2), T, 0, stream>>>(h1, x1, hcat, n);
    k_wmma_gemm<128><<<gg, 128, 0, stream>>>(hcat, w2, t2, n, K2);
    k_agg_init<<<g1((size_t)n * 16), T, 0, stream>>>(t2, dinv, b2, h2, n);
    k_agg_edges<<<g1((size_t)E * 16), T, 0, stream>>>(edge, edge + E, dinv, t2, h2, E);

    // elu -> fc -> mask -> log_softmax over axis 0
    k_logits<<<g1(n), T, 0, stream>>>(h2, fcw, fcb, emask, logits, n);
    k_max<<<1, 256, 0, stream>>>(logits, n, red);
    k_sumexp<<<1, 256, 0, stream>>>(logits, n, red);
    k_final<<<g1(n), T, 0, stream>>>(logits, red, out, n);
}